// NLRidge_78383153152463
// MI455X (gfx1250) — compile-verified
//
#include <hip/hip_runtime.h>

// ---------------------------------------------------------------------------
// NL-Ridge denoiser for MI455X (gfx1250, wave32, WMMA + TDM).
//
// Pipeline (matches the JAX reference, N=C=1, H=W=256):
//   pad(reflect,21) -> block_match(m=18) -> denoise1 -> agg -> den1
//   pad(den1)       -> block_match(m=55) -> denoise2 -> agg -> out
//
// Algebra: theta = I - c*G^{-1} with G = YtY (stage1) / XtX + cI (stage2),
// c = 49*sigma^2.  One workgroup (256 thr = 8 wave32) per patch group:
//   gather -> Gram via V_WMMA_F32_16X16X4_F32 -> Gauss-Jordan inverse in LDS
//   -> Xh = theta@Y via WMMA -> atomic aggregation into 298x298 num/den.
//
// Block matching: all 1369 candidate patches for one reference position lie
// in a 43x43 window (7.4 KB) of the padded image.  That window is DMA'd into
// LDS once per workgroup with TENSOR_LOAD_TO_LDS (TDM 2D tile), waited with
// s_wait_tensorcnt, then all SSDs are computed from LDS -- ~150x less VMEM
// traffic than per-offset global reads.  Fallback path: plain copy loop.
// ---------------------------------------------------------------------------

#define HH     256
#define PAD    21
#define HP     298              // HH + 2*PAD
#define GRID   292              // HP - 7 + 1
#define WIN    37
#define NOFF   (WIN * WIN)      // 1369
#define CENTER (18 * WIN + 18)  // 684 (forced first pick)
#define PATCH  7
#define PP     49
#define NB     4096             // 64x64 reference positions (stride 4)
#define NPC    64               // patch-pixel dim padded 49 -> 64
#define TS     43               // search-window tile side (37 + 7 - 1)

typedef __attribute__((ext_vector_type(2))) float    v2f;
typedef __attribute__((ext_vector_type(8))) float    v8f;
typedef __attribute__((ext_vector_type(4))) unsigned u32x4;
typedef __attribute__((ext_vector_type(4))) int      i32x4;
typedef __attribute__((ext_vector_type(8))) int      i32x8;

#if defined(__AMDGCN__) && __has_builtin(__builtin_amdgcn_wmma_f32_16x16x4_f32)
#define USE_WMMA 1
#else
#define USE_WMMA 0
#endif

#if defined(__AMDGCN__) && __has_builtin(__builtin_amdgcn_tensor_load_to_lds) && \
    __has_builtin(__builtin_amdgcn_s_wait_tensorcnt)
#define USE_TDM 1
#else
#define USE_TDM 0
#endif

// ---------------------------------------------------------------- utilities
__global__ __launch_bounds__(256) void zero_kernel(float* __restrict__ p, int n) {
    int t = blockIdx.x * blockDim.x + threadIdx.x;
    if (t < n) p[t] = 0.0f;
}

__global__ __launch_bounds__(256) void pad_reflect_kernel(const float* __restrict__ src,
                                                          float* __restrict__ dst) {
    int t = blockIdx.x * blockDim.x + threadIdx.x;
    if (t >= HP * HP) return;
    int y = t / HP, x = t % HP;
    int sy = y - PAD, sx = x - PAD;
    sy = (sy < 0) ? -sy : ((sy >= HH) ? (2 * HH - 2 - sy) : sy);
    sx = (sx < 0) ? -sx : ((sx >= HH) ? (2 * HH - 2 - sx) : sx);
    dst[t] = src[sy * HH + sx];
}

__global__ __launch_bounds__(256) void finalize_kernel(const float* __restrict__ num,
                                                       const float* __restrict__ den,
                                                       float* __restrict__ out) {
    int t = blockIdx.x * blockDim.x + threadIdx.x;
    if (t >= HH * HH) return;
    int y = t / HH, x = t % HH;
    int a = (y + PAD) * HP + (x + PAD);
    out[t] = num[a] / den[a];
}

// ------------------------------------------------------------ block matching
// One workgroup per reference position.  The 43x43 search window is brought
// into LDS via the Tensor Data Mover (one 2D tile descriptor, TENSORcnt),
// then 1369 SSDs are computed from LDS, then M rounds of parallel argmin
// (ties -> lowest index, matching lax.top_k; center offset forced first).
template <int M>
__global__ __launch_bounds__(256)
void block_match_kernel(const float* __restrict__ img, int* __restrict__ idx_out) {
    __shared__ float tile[TS * TS];   // 43x43 search window
    __shared__ float dist[NOFF];
    __shared__ float rpatch[PP];
    __shared__ float redv[256];
    __shared__ int   redi[256];

    const int tid = threadIdx.x, bd = blockDim.x;
    const int b = blockIdx.x;
    const int R = 18 + 4 * (b / 64);        // reference patch top-left (padded img)
    const int C = 18 + 4 * (b % 64);
    const float* wsrc = img + (R - 18) * HP + (C - 18);   // window origin

#if USE_TDM
    if (tid < 32) {   // one wave issues the DMA (EXEC-independent)
        // D# group 0: count=1 | lds_addr | global_addr(57b) | type=2
        unsigned lds_addr = (unsigned)(unsigned long long)(tile);  // flat[31:0] == LDS offset
        unsigned long long ga = (unsigned long long)(wsrc);
        u32x4 g0;
        g0.x = 1u;                                  // count=1, user mode
        g0.y = lds_addr;                            // bits 63:32  lds_addr
        g0.z = (unsigned)(ga & 0xFFFFFFFFu);        // bits 95:64  global_addr lo
        g0.w = (unsigned)((ga >> 32) & 0x1FFFFFFu)  // bits 120:96 global_addr hi
             | (2u << 30);                          // bits 127:126 type=2 (image)
        // D# group 1: data_size=4B, tensor 298x298, tile 43x43, stride 298
        i32x8 g1;
        g1[0] = (2 << 16);                          // data_size=2 -> 4 bytes
        g1[1] = (HP & 0xFFFF) << 16;                // tensor_dim0[15:0] @ bit48
        g1[2] = (HP & 0xFFFF) << 16;                // dim0 hi=0 | tensor_dim1[15:0] @ bit80
        g1[3] = (TS & 0xFFFF) << 16;                // dim1 hi=0 | tile_dim0 @ bit112
        g1[4] = (TS & 0xFFFF);                      // tile_dim1 @ bit128, tile_dim2=0
        g1[5] = HP;                                 // tensor_dim0_stride[31:0] @ bit160
        g1[6] = 0;                                  // stride hi / tensor_dim1_stride lo
        g1[7] = 0;
        i32x4 gz = {0, 0, 0, 0};                    // groups 2/3 unused (2D tensor)
#if __clang_major__ >= 23
        i32x8 gz8 = {0, 0, 0, 0, 0, 0, 0, 0};
        __builtin_amdgcn_tensor_load_to_lds(g0, g1, gz, gz, gz8, 0);
#else
        __builtin_amdgcn_tensor_load_to_lds(g0, g1, gz, gz, 0);
#endif
        __builtin_amdgcn_s_wait_tensorcnt(0);
    }
#else
    for (int t = tid; t < TS * TS; t += bd) {
        const float* p = wsrc + (t / TS) * HP + (t % TS);
        __builtin_prefetch(p + HP, 0, 0);           // global_prefetch_b8
        tile[t] = *p;
    }
#endif
    __syncthreads();

    // reference patch lives at tile(18,18)
    if (tid < PP) rpatch[tid] = tile[(18 + tid / PATCH) * TS + (18 + tid % PATCH)];
    __syncthreads();

    for (int o = tid; o < NOFF; o += bd) {
        const float* p = &tile[(o / WIN) * TS + (o % WIN)];
        float s = 0.0f;
#pragma unroll
        for (int u = 0; u < PATCH; ++u)
#pragma unroll
            for (int v = 0; v < PATCH; ++v) {
                float d = p[u * TS + v] - rpatch[u * PATCH + v];
                s += d * d;
            }
        dist[o] = s;
    }
    __syncthreads();
    if (tid == 0) dist[CENTER] = -1.0f;   // center always selected first
    __syncthreads();

    for (int t = 0; t < M; ++t) {
        float bv = 3.4e38f;
        int   bi = NOFF;
        for (int o = tid; o < NOFF; o += bd) {
            float v = dist[o];
            if (v < bv || (v == bv && o < bi)) { bv = v; bi = o; }
        }
        redv[tid] = bv; redi[tid] = bi;
        __syncthreads();
        for (int s = 128; s > 0; s >>= 1) {
            if (tid < s) {
                float ov = redv[tid + s]; int oi = redi[tid + s];
                if (ov < redv[tid] || (ov == redv[tid] && oi < redi[tid])) {
                    redv[tid] = ov; redi[tid] = oi;
                }
            }
            __syncthreads();
        }
        if (tid == 0) {
            int w = redi[0];
            int pr = R + w / WIN - 18;
            int pc = C + w % WIN - 18;
            idx_out[b * M + t] = pr * GRID + pc;
            dist[w] = 3.4e38f;
        }
        __syncthreads();
    }
}

// ------------------------------------------------------------- fused denoise
// M = group size (18 / 55), MP = padded to WMMA multiple (32 / 64).
// gimg feeds the Gram (noisy stage1 / pilot stage2), yimg feeds theta@Y.
template <int M, int MP>
__global__ __launch_bounds__(256)
void denoise_kernel(const float* __restrict__ gimg, const float* __restrict__ yimg,
                    const int* __restrict__ idxb, const float* __restrict__ sigp,
                    float* __restrict__ num, float* __restrict__ den, float addDiag) {
    __shared__ float Xs[MP * NPC];   // Gram-source patches, later reused for theta
    __shared__ float Ys[MP * NPC];   // noisy patches
    __shared__ float Am[MP * MP];    // Gram -> in-place inverse
    __shared__ float fcol[MP];
    __shared__ float wrow[MP];
    __shared__ int   posr[MP];
    __shared__ float s_piv;

    const int tid = threadIdx.x, bd = blockDim.x;
    const int b = blockIdx.x;
    const int lane = tid & 31, wave = tid >> 5, nw = bd >> 5;
    const float sig = sigp[0];
    const float creg = (float)PP * sig * sig;

    if (tid < MP) posr[tid] = (tid < M) ? idxb[b * M + tid] : 0;
    __syncthreads();

    // ---- gather patches (zero-padded to MP x 64)
    for (int t = tid; t < MP * NPC; t += bd) {
        int i = t >> 6, k = t & 63;
        float xv = 0.0f, yv = 0.0f;
        if (i < M && k < PP) {
            int pos = posr[i];
            int pr = pos / GRID, pc = pos % GRID;
            int a = (pr + k / PATCH) * HP + (pc + k % PATCH);
            xv = gimg[a];
            yv = yimg[a];
        }
        Xs[t] = xv; Ys[t] = yv;
    }
    __syncthreads();

    // ---- Gram: Am = Xs * Xs^T   (f32 WMMA 16x16x4 tiles)
    constexpr int T = MP / 16;
#if USE_WMMA
    {
        const int lr = lane & 15, lk = (lane >> 4) << 1;
        for (int tile = wave; tile < T * T; tile += nw) {
            int ti = tile / T, tj = tile % T;
            v8f acc = {0.f, 0.f, 0.f, 0.f, 0.f, 0.f, 0.f, 0.f};
            for (int ks = 0; ks < NPC; ks += 4) {
                v2f a, bb;
                a.x  = Xs[(ti * 16 + lr) * NPC + ks + lk];
                a.y  = Xs[(ti * 16 + lr) * NPC + ks + lk + 1];
                bb.x = Xs[(tj * 16 + lr) * NPC + ks + lk];
                bb.y = Xs[(tj * 16 + lr) * NPC + ks + lk + 1];
                acc = __builtin_amdgcn_wmma_f32_16x16x4_f32(
                    false, a, false, bb, (short)0, acc, false, false);
            }
            int row0 = ti * 16 + ((lane >> 4) << 3);
            int col  = tj * 16 + lr;
#pragma unroll
            for (int r = 0; r < 8; ++r) Am[(row0 + r) * MP + col] = acc[r];
        }
    }
#else
    for (int t = tid; t < MP * MP; t += bd) {
        int i = t / MP, j = t % MP;
        float s = 0.0f;
        for (int k = 0; k < NPC; ++k) s += Xs[i * NPC + k] * Xs[j * NPC + k];
        Am[t] = s;
    }
#endif
    __syncthreads();

    // ---- diag fixup: + c (stage2) on real rows, identity on pad rows
    if (tid < MP) {
        if (tid < M) Am[tid * MP + tid] += addDiag * creg;
        else         Am[tid * MP + tid] = 1.0f;
    }
    __syncthreads();

    // ---- in-place Gauss-Jordan inversion (SPD, no pivoting)
    for (int k = 0; k < MP; ++k) {
        if (tid == 0) s_piv = 1.0f / Am[k * MP + k];
        if (tid < MP) fcol[tid] = Am[tid * MP + k];
        __syncthreads();
        float pv = s_piv;
        for (int j = tid; j < MP; j += bd)
            Am[k * MP + j] = (j == k) ? pv : Am[k * MP + j] * pv;
        __syncthreads();
        for (int t = tid; t < MP * MP; t += bd) {
            int r = t / MP, j = t % MP;
            if (r != k) {
                float f = fcol[r];
                Am[t] = (j == k) ? (-f * pv) : (Am[t] - f * Am[k * MP + j]);
            }
        }
        __syncthreads();
    }

    // ---- theta = I - c * G^{-1}   (reuse Xs as theta, stride MP)
    float* th = Xs;
    for (int t = tid; t < MP * MP; t += bd) {
        int i = t / MP, j = t % MP;
        th[i * MP + j] = ((i == j) ? 1.0f : 0.0f) - creg * Am[t];
    }
    __syncthreads();
    if (tid < MP) {
        float s = 0.0f;
        for (int j = 0; j < MP; ++j) { float v = th[tid * MP + j]; s += v * v; }
        wrow[tid] = 1.0f / s;
    }
    __syncthreads();

    // ---- Xh = theta @ Y, weighted atomic aggregation
#if USE_WMMA
    {
        const int lr = lane & 15, lk = (lane >> 4) << 1;
        constexpr int CT = NPC / 16;   // 4 column tiles
        for (int tile = wave; tile < T * CT; tile += nw) {
            int ti = tile / CT, tj = tile % CT;
            v8f acc = {0.f, 0.f, 0.f, 0.f, 0.f, 0.f, 0.f, 0.f};
            for (int ks = 0; ks < MP; ks += 4) {
                v2f a, bb;
                a.x  = th[(ti * 16 + lr) * MP + ks + lk];
                a.y  = th[(ti * 16 + lr) * MP + ks + lk + 1];
                bb.x = Ys[(ks + lk) * NPC + tj * 16 + lr];
                bb.y = Ys[(ks + lk + 1) * NPC + tj * 16 + lr];
                acc = __builtin_amdgcn_wmma_f32_16x16x4_f32(
                    false, a, false, bb, (short)0, acc, false, false);
            }
            int row0 = ti * 16 + ((lane >> 4) << 3);
            int col  = tj * 16 + lr;
#pragma unroll
            for (int r = 0; r < 8; ++r) {
                int row = row0 + r;
                if (row < M && col < PP) {
                    int pos = posr[row];
                    int pr = pos / GRID, pc = pos % GRID;
                    int a2 = (pr + col / PATCH) * HP + (pc + col % PATCH);
                    float w = wrow[row];
                    atomicAdd(&num[a2], w * acc[r]);
                    atomicAdd(&den[a2], w);
                }
            }
        }
    }
#else
    for (int t = tid; t < M * PP; t += bd) {
        int i = t / PP, k = t % PP;
        float s = 0.0f;
        for (int j = 0; j < MP; ++j) s += th[i * MP + j] * Ys[j * NPC + k];
        int pos = posr[i];
        int pr = pos / GRID, pc = pos % GRID;
        int a2 = (pr + k / PATCH) * HP + (pc + k % PATCH);
        float w = wrow[i];
        atomicAdd(&num[a2], w * s);
        atomicAdd(&den[a2], w);
    }
#endif
}

// --------------------------------------------------------------- launch glue
extern "C" void kernel_launch(void* const* d_in, const int* in_sizes, int n_in,
                              void* d_out, int out_size, void* d_ws, size_t ws_size,
                              hipStream_t stream) {
    (void)in_sizes; (void)n_in; (void)out_size; (void)ws_size;
    const float* input = (const float*)d_in[0];
    const float* sigma = (const float*)d_in[1];
    float* out = (float*)d_out;

    // workspace carve-out (~3 MB of floats/ints)
    float* wsf = (float*)d_ws;
    size_t o = 0;
    auto alloc = [&](size_t n) { float* p = wsf + o; o += (n + 63) & ~(size_t)63; return p; };
    float* ypad = alloc((size_t)HP * HP);
    float* xpad = alloc((size_t)HP * HP);
    float* numb = alloc((size_t)HP * HP);
    float* denb = alloc((size_t)HP * HP);
    float* den1 = alloc((size_t)HH * HH);
    int*   idx1 = (int*)alloc((size_t)NB * 18);
    int*   idx2 = (int*)alloc((size_t)NB * 55);

    const dim3 blk(256);
    const int npad = HP * HP;
    const int gpad = (npad + 255) / 256;
    const int gimg = (HH * HH + 255) / 256;

    // ---- stage 1 (p=7, m=18)
    pad_reflect_kernel<<<gpad, blk, 0, stream>>>(input, ypad);
    zero_kernel<<<gpad, blk, 0, stream>>>(numb, npad);
    zero_kernel<<<gpad, blk, 0, stream>>>(denb, npad);
    block_match_kernel<18><<<NB, blk, 0, stream>>>(ypad, idx1);
    denoise_kernel<18, 32><<<NB, blk, 0, stream>>>(ypad, ypad, idx1, sigma,
                                                   numb, denb, 0.0f);
    finalize_kernel<<<gimg, blk, 0, stream>>>(numb, denb, den1);

    // ---- stage 2 (p=7, m=55), pilot = den1
    pad_reflect_kernel<<<gpad, blk, 0, stream>>>(den1, xpad);
    zero_kernel<<<gpad, blk, 0, stream>>>(numb, npad);
    zero_kernel<<<gpad, blk, 0, stream>>>(denb, npad);
    block_match_kernel<55><<<NB, blk, 0, stream>>>(xpad, idx2);
    denoise_kernel<55, 64><<<NB, blk, 0, stream>>>(xpad, ypad, idx2, sigma,
                                                   numb, denb, 1.0f);
    finalize_kernel<<<gimg, blk, 0, stream>>>(numb, denb, out);
}